// MixerBlock_15169824489515
// MI455X (gfx1250) — compile-verified
//
#include <hip/hip_runtime.h>
#include <math.h>

// MI455X (gfx1250) fused MLP-Mixer block, fp32 via V_WMMA_F32_16X16X4_F32.
// Memory-bound (275 GFLOP vs ~0.5 GB traffic @ 23.3 TB/s): fp32 matrix path
// matches reference precision; compute ~1us vs ~30us HBM. This revision fixes
// the transposed-store epilogue (LDS-staged, coalesced, no 32x cacheline
// amplification) and adds next-chunk global_prefetch.

typedef __attribute__((ext_vector_type(2))) float v2f;
typedef __attribute__((ext_vector_type(8))) float v8f;

#define LN_EPS 1e-5f
#define BM 64
#define BN 64
#define BK 32
#define LDSP 65   // padded LDS row stride (odd -> conflict-free columns)

__device__ __forceinline__ float gelu_exact(float x) {
    return 0.5f * x * (1.0f + erff(x * 0.70710678118654752440f));
}

// LayerNorm over last dim H (=1024). One block of 256 threads per row.
__global__ __launch_bounds__(256)
void ln_kernel(const float* __restrict__ x, const float* __restrict__ g,
               const float* __restrict__ b, float* __restrict__ y, int H) {
    __shared__ float s1[256];
    __shared__ float s2[256];
    const int r   = blockIdx.x;
    const int tid = threadIdx.x;
    const float* xr = x + (long long)r * H;
    float v[4];
    float sum = 0.f, sq = 0.f;
#pragma unroll
    for (int i = 0; i < 4; ++i) {
        v[i] = xr[tid + i * 256];
        sum += v[i];
        sq  += v[i] * v[i];
    }
    s1[tid] = sum;
    s2[tid] = sq;
    __syncthreads();
    for (int off = 128; off > 0; off >>= 1) {
        if (tid < off) { s1[tid] += s1[tid + off]; s2[tid] += s2[tid + off]; }
        __syncthreads();
    }
    const float mean = s1[0] / (float)H;
    const float var  = s2[0] / (float)H - mean * mean;
    const float inv  = rsqrtf(var + LN_EPS);
    float* yr = y + (long long)r * H;
#pragma unroll
    for (int i = 0; i < 4; ++i) {
        const int h = tid + i * 256;
        yr[h] = (v[i] - mean) * inv * g[h] + b[h];
    }
}

// Tiled WMMA GEMM: C[M,N] = epilogue( A[M,K] * Bw[N,K]^T + bias[N] )
//   AMODE 0: A element at m*lda + k     (row-major M x K)
//   AMODE 1: A element at k*lda + m     (transposed source, coalesced over m)
//   MASKB 1: causal mask on Bw (k > n zeroed); K-loop truncated at nBlock+BN.
//   EPI 0: store gelu(val) to C[m*ldc + n]
//   EPI 1: store R[m*ldr + n] + val to C[m*ldc + n]   (residual, in-place OK)
//   EPI 2: store R[n*ldr + m] + val to C[n*ldc + m]   (LDS-staged transpose)
template <int AMODE, int MASKB, int EPI>
__global__ __launch_bounds__(128)
void gemm_wmma(const float* __restrict__ A, long long aBatch, int lda,
               const float* __restrict__ Bw, int ldb,
               const float* __restrict__ bias,
               const float* __restrict__ R, long long rBatch, int ldr,
               float* __restrict__ C, long long cBatch, int ldc, int K) {
    // 4160 floats: [32][65] A-stage + [32][65] B-stage, aliased as a
    // [64][65] output-transpose stage after the K-loop completes.
    __shared__ __align__(16) float smem[2 * BK * LDSP];
    float (*As)[LDSP] = (float (*)[LDSP])smem;
    float (*Bs)[LDSP] = (float (*)[LDSP])(smem + BK * LDSP);
    float (*Ct)[LDSP] = (float (*)[LDSP])smem;   // epilogue alias (64x65)

    const int tid  = threadIdx.x;
    const int wave = tid >> 5;          // wave32
    const int lane = tid & 31;
    const int half = lane >> 4;         // 0: lanes 0-15, 1: lanes 16-31
    const int lidx = lane & 15;

    const int mBlock = blockIdx.x * BM;
    const int nBlock = blockIdx.y * BN;
    const int batch  = blockIdx.z;

    const float* Ab = A + (long long)batch * aBatch;
    const float* Rb = (EPI != 0) ? (R + (long long)batch * rBatch) : A;
    float*       Cb = C + (long long)batch * cBatch;

    const int wm = (wave >> 1) * 32;    // wave's 32x32 region of the 64x64 tile
    const int wn = (wave & 1) * 32;

    v8f acc[2][2];
#pragma unroll
    for (int i = 0; i < 2; ++i)
#pragma unroll
        for (int j = 0; j < 2; ++j) {
            v8f z = {0.f, 0.f, 0.f, 0.f, 0.f, 0.f, 0.f, 0.f};
            acc[i][j] = z;
        }

    int kEnd = K;
    if (MASKB) {
        const int lim = nBlock + BN;
        kEnd = lim < K ? lim : K;
    }

    for (int k0 = 0; k0 < kEnd; k0 += BK) {
        // ---- stage A tile into LDS as As[k][m] ----
        if (AMODE == 0) {
#pragma unroll
            for (int it = 0; it < (BM * BK) / 128; ++it) {
                const int idx = tid + it * 128;
                const int m = idx >> 5;       // / BK
                const int k = idx & (BK - 1); // coalesced over k
                As[k][m] = Ab[(long long)(mBlock + m) * lda + (k0 + k)];
            }
        } else {
#pragma unroll
            for (int it = 0; it < (BM * BK) / 128; ++it) {
                const int idx = tid + it * 128;
                const int k = idx >> 6;       // / BM
                const int m = idx & (BM - 1); // coalesced over m
                As[k][m] = Ab[(long long)(k0 + k) * lda + (mBlock + m)];
            }
        }
        // ---- stage B tile into LDS as Bs[k][n], applying causal mask ----
#pragma unroll
        for (int it = 0; it < (BN * BK) / 128; ++it) {
            const int idx = tid + it * 128;
            const int n = idx >> 5;
            const int k = idx & (BK - 1);
            const int ng = nBlock + n;
            const int kg = k0 + k;
            float w = Bw[(long long)ng * ldb + kg];
            if (MASKB && (kg > ng)) w = 0.f;
            Bs[k][n] = w;
        }
        // ---- prefetch next K-chunk (uniform branch; global_prefetch_b8) ----
        if (k0 + BK < kEnd) {
            if (AMODE == 0) {
                __builtin_prefetch(
                    &Ab[(long long)(mBlock + (tid >> 5)) * lda +
                        (k0 + BK + (tid & 31))], 0, 3);
            } else {
                __builtin_prefetch(
                    &Ab[(long long)(k0 + BK + (tid >> 6)) * lda +
                        (mBlock + (tid & 63))], 0, 3);
            }
            __builtin_prefetch(
                &Bw[(long long)(nBlock + (tid >> 5)) * ldb +
                    (k0 + BK + (tid & 31))], 0, 3);
        }
        __syncthreads();

        // ---- 8 k-steps of 4 WMMAs (2x2 tile grid per wave) ----
#pragma unroll
        for (int kk = 0; kk < BK; kk += 4) {
            v2f a[2], bf[2];
#pragma unroll
            for (int i = 0; i < 2; ++i) {
                // A 16x4 f32: lanes 0-15 hold K={kk,kk+1}, 16-31 K={kk+2,kk+3}
                a[i].x = As[kk + 2 * half + 0][wm + i * 16 + lidx];
                a[i].y = As[kk + 2 * half + 1][wm + i * 16 + lidx];
            }
#pragma unroll
            for (int j = 0; j < 2; ++j) {
                bf[j].x = Bs[kk + 2 * half + 0][wn + j * 16 + lidx];
                bf[j].y = Bs[kk + 2 * half + 1][wn + j * 16 + lidx];
            }
#pragma unroll
            for (int i = 0; i < 2; ++i)
#pragma unroll
                for (int j = 0; j < 2; ++j)
                    acc[i][j] = __builtin_amdgcn_wmma_f32_16x16x4_f32(
                        false, a[i], false, bf[j], (short)0, acc[i][j],
                        false, false);
        }
        __syncthreads();
    }

    // ---- epilogue: C/D 16x16 f32 layout: lane -> N, vgpr v -> M = half*8+v
    if (EPI == 2) {
        // Stage biased tile into LDS (aliases As/Bs; safe after final sync),
        // then cooperative transposed + residual store, fully coalesced.
#pragma unroll
        for (int i = 0; i < 2; ++i) {
#pragma unroll
            for (int j = 0; j < 2; ++j) {
                const int Nl = wn + j * 16 + lidx;
                const float bn = bias[nBlock + Nl];
#pragma unroll
                for (int v = 0; v < 8; ++v) {
                    const int Ml = wm + i * 16 + half * 8 + v;
                    Ct[Ml][Nl] = acc[i][j][v] + bn;
                }
            }
        }
        __syncthreads();
#pragma unroll
        for (int it = 0; it < (BM * BN) / 128; ++it) {
            const int idx = tid + it * 128;
            const int m = idx & (BM - 1);   // fast dim -> coalesced over m
            const int n = idx >> 6;
            Cb[(long long)(nBlock + n) * ldc + (mBlock + m)] =
                Rb[(long long)(nBlock + n) * ldr + (mBlock + m)] + Ct[m][n];
        }
    } else {
#pragma unroll
        for (int i = 0; i < 2; ++i) {
#pragma unroll
            for (int j = 0; j < 2; ++j) {
                const int Ng = nBlock + wn + j * 16 + lidx;
                const float bn = bias[Ng];
#pragma unroll
                for (int v = 0; v < 8; ++v) {
                    const int Mg = mBlock + wm + i * 16 + half * 8 + v;
                    const float val = acc[i][j][v] + bn;
                    if (EPI == 0) {
                        Cb[(long long)Mg * ldc + Ng] = gelu_exact(val);
                    } else {
                        Cb[(long long)Mg * ldc + Ng] =
                            Rb[(long long)Mg * ldr + Ng] + val;
                    }
                }
            }
        }
    }
}

extern "C" void kernel_launch(void* const* d_in, const int* in_sizes, int n_in,
                              void* d_out, int out_size, void* d_ws,
                              size_t ws_size, hipStream_t stream) {
    (void)in_sizes; (void)n_in; (void)out_size; (void)ws_size;
    const int B = 32, T = 1024, H = 1024;

    const float* x    = (const float*)d_in[0];
    const float* tw1  = (const float*)d_in[1];
    const float* tb1  = (const float*)d_in[2];
    const float* tw2  = (const float*)d_in[3];
    const float* tb2  = (const float*)d_in[4];
    const float* cw1  = (const float*)d_in[5];
    const float* cb1  = (const float*)d_in[6];
    const float* cw2  = (const float*)d_in[7];
    const float* cb2  = (const float*)d_in[8];
    const float* ln1g = (const float*)d_in[9];
    const float* ln1b = (const float*)d_in[10];
    const float* ln2g = (const float*)d_in[11];
    const float* ln2b = (const float*)d_in[12];
    float* out = (float*)d_out;

    float* ws0 = (float*)d_ws;                       // 128 MB scratch A
    float* ws1 = ws0 + (size_t)B * T * H;            // 128 MB scratch B
    const long long BTH = (long long)T * H;

    // 1) y = LN1(x)                        -> ws0 [B,T,H]
    ln_kernel<<<B * T, 256, 0, stream>>>(x, ln1g, ln1b, ws0, H);

    // 2) u = gelu( y^T @ (mask*tw1)^T + tb1 )  per batch, M=h,N=t_out,K=t_in
    //    A read transposed from [B,T,H]    -> ws1 [B,H,T]
    gemm_wmma<1, 1, 0><<<dim3(H / BM, T / BN, B), 128, 0, stream>>>(
        ws0, BTH, H, tw1, T, tb1, nullptr, 0, 0, ws1, BTH, T, T);

    // 3) x1 = x + ( u @ (mask*tw2)^T + tb2 )^T -> d_out [B,T,H]
    gemm_wmma<0, 1, 2><<<dim3(H / BM, T / BN, B), 128, 0, stream>>>(
        ws1, BTH, T, tw2, T, tb2, x, BTH, H, out, BTH, H, T);

    // 4) z = LN2(x1)                       -> ws0 [B*T, H]
    ln_kernel<<<B * T, 256, 0, stream>>>(out, ln2g, ln2b, ws0, H);

    // 5) h = gelu( z @ cw1^T + cb1 )       -> ws1 [B*T, H]
    gemm_wmma<0, 0, 0><<<dim3((B * T) / BM, H / BN, 1), 128, 0, stream>>>(
        ws0, 0, H, cw1, H, cb1, nullptr, 0, 0, ws1, 0, H, H);

    // 6) out = x1 + h @ cw2^T + cb2        -> d_out (in-place RMW, per-thread)
    gemm_wmma<0, 0, 1><<<dim3((B * T) / BM, H / BN, 1), 128, 0, stream>>>(
        ws1, 0, H, cw2, H, cb2, out, 0, H, out, 0, H, H);
}